// MixProp1_83975200572031
// MI455X (gfx1250) — compile-verified
//
#include <hip/hip_runtime.h>

// Problem constants (from reference)
#define NN    4096   // graph nodes
#define TT    32     // time steps
#define CC    32     // input channels
#define BB    8      // batch
#define COUT  32     // output channels
#define KTOT  96     // (GDEP+1)*CC
#define ALPHA 0.05f

typedef __attribute__((ext_vector_type(2))) float v2f;
typedef __attribute__((ext_vector_type(8))) float v8f;

__device__ __forceinline__ v8f wmma_f32(v2f a, v2f b, v8f c) {
  // D(16x16,f32) = A(16x4,f32) x B(4x16,f32) + C
  return __builtin_amdgcn_wmma_f32_16x16x4_f32(
      /*neg_a=*/false, a, /*neg_b=*/false, b,
      /*c_mod=*/(short)0, c, /*reuse_a=*/false, /*reuse_b=*/false);
}

// gfx1250 async copies global -> LDS (ASYNCcnt tracked, complete in-order)
__device__ __forceinline__ void async_copy_b128(unsigned lds_addr, const float* gaddr) {
  asm volatile("global_load_async_to_lds_b128 %0, %1, off"
               :: "v"(lds_addr), "v"(gaddr) : "memory");
}
__device__ __forceinline__ void async_copy_b32(unsigned lds_addr, const float* gaddr) {
  asm volatile("global_load_async_to_lds_b32 %0, %1, off"
               :: "v"(lds_addr), "v"(gaddr) : "memory");
}

// ---------------------------------------------------------------------------
// Kernel 1: invd[w] = 1 / (sum_j adj[w][j] + 1)   (row sums of adj + I)
// ---------------------------------------------------------------------------
__global__ __launch_bounds__(256) void MixProp_deg(const float* __restrict__ adj,
                                                   float* __restrict__ invd) {
  __shared__ float red[256];
  const int row = blockIdx.x;
  const int tid = threadIdx.x;
  float p = 0.0f;
  const float* r = adj + (size_t)row * NN;
  for (int j = tid; j < NN; j += 256) p += r[j];
  red[tid] = p;
  __syncthreads();
  for (int s = 128; s > 0; s >>= 1) {
    if (tid < s) red[tid] += red[tid + s];
    __syncthreads();
  }
  if (tid == 0) invd[row] = 1.0f / (red[0] + 1.0f);
}

// ---------------------------------------------------------------------------
// Kernel 2: anorm[v,w] = (adj[v,w] + (v==w)) * invd[w]
// ---------------------------------------------------------------------------
__global__ __launch_bounds__(256) void MixProp_norm(const float* __restrict__ adj,
                                                    const float* __restrict__ invd,
                                                    float* __restrict__ anorm) {
  const size_t q = (size_t)blockIdx.x * 256 + threadIdx.x;   // float4 index
  const size_t e = q * 4;
  const int v = (int)(e >> 12);          // e / 4096
  const int w = (int)(e & (NN - 1));     // e % 4096 (multiple of 4)
  float4 av = *reinterpret_cast<const float4*>(adj + e);
  const float4 dv = *reinterpret_cast<const float4*>(invd + w);
  const int d = v - w;
  if ((unsigned)d < 4u) (&av.x)[d] += 1.0f;    // + identity
  av.x *= dv.x; av.y *= dv.y; av.z *= dv.z; av.w *= dv.w;
  *reinterpret_cast<float4*>(anorm + e) = av;
}

// ---------------------------------------------------------------------------
// Kernel 3: dst[b,c,v,t] = ALPHA*x[b,c,v,t] + sum_w anorm[v,w] * src[b,c,w,t]
// WG = 256 thr (8 waves). Output tile: 64 rows (v) x 128 cols (4 chan x 32 t).
// Double-buffered LDS, async-fed: A via B128 (contiguous both sides),
// H via B32 (coalesced global t, transposed scatter into [col][k] so fragment
// reads are single aligned b64 loads). 18 async ops/wave/chunk.
// ---------------------------------------------------------------------------
#define ASTR 36    // padded LDS row stride (floats): conflict-free fragment reads
__global__ __launch_bounds__(256) void MixProp_gemm(const float* __restrict__ anorm,
                                                    const float* __restrict__ src,
                                                    const float* __restrict__ x,
                                                    float* __restrict__ dst) {
  __shared__ __align__(16) float As[2][64 * ASTR];    // [r][k], k<32
  __shared__ __align__(16) float Hs[2][128 * ASTR];   // [col][k], col<128

  const int tid  = threadIdx.x;
  const int lane = tid & 31;
  const int wid  = tid >> 5;
  const int l16  = lane & 15;
  const int half = lane >> 4;          // K-pair select per ISA A/B layout

  const int v0 = blockIdx.x * 64;      // node-tile base
  const int b  = blockIdx.y;           // batch
  const int bc = b * CC + blockIdx.z * 4;  // first of 4 channels

  const int msub = wid & 3;            // 16-row subtile
  const int colw = (wid >> 2) * 64;    // wave column strip base

  v8f acc[4];
  #pragma unroll
  for (int j = 0; j < 4; ++j) acc[j] = (v8f){0,0,0,0,0,0,0,0};

  // stage one K-chunk (32) into buffer p
  auto stage = [&](int ko, int p) {
    // A tile 64x32: 2 x B128 per thread (16B contiguous in global and LDS)
    #pragma unroll
    for (int i = 0; i < 2; ++i) {
      const int e4 = tid + (i << 8);
      const int r  = e4 >> 3;
      const int k4 = (e4 & 7) << 2;
      async_copy_b128((unsigned)(size_t)&As[p][r * ASTR + k4],
                      anorm + (size_t)(v0 + r) * NN + ko + k4);
    }
    // H tile 32(k) x 128(col): 16 x B32 per thread, k<->t transpose at LDS write.
    // Wave-consecutive lanes -> consecutive t: 128B coalesced global requests.
    #pragma unroll
    for (int i = 0; i < 16; ++i) {
      const int e  = tid + (i << 8);
      const int t  = e & 31;
      const int k  = (e >> 5) & 31;
      const int cl = e >> 10;          // channel 0..3
      async_copy_b32((unsigned)(size_t)&Hs[p][(cl * 32 + t) * ASTR + k],
                     src + ((size_t)(bc + cl) * NN + ko + k) * TT + t);
    }
  };

  stage(0, 0);
  int p = 0;
  for (int ko = 0; ko < NN; ko += 32) {
    const bool more = (ko + 32) < NN;
    if (more) stage(ko + 32, p ^ 1);               // prefetch next chunk (other buffer)
    if (more) asm volatile("s_wait_asynccnt 0x12" ::: "memory"); // cur buf resident
    else      asm volatile("s_wait_asynccnt 0x0"  ::: "memory");
    __syncthreads();

    const float* Ac = &As[p][0];
    const float* Hc = &Hs[p][0];
    #pragma unroll
    for (int ks = 0; ks < 32; ks += 4) {
      v2f a;
      a.x = Ac[(msub * 16 + l16) * ASTR + ks + 2 * half];
      a.y = Ac[(msub * 16 + l16) * ASTR + ks + 2 * half + 1];
      #pragma unroll
      for (int j = 0; j < 4; ++j) {
        const int col = colw + j * 16 + l16;
        v2f bb;
        bb.x = Hc[col * ASTR + ks + 2 * half];
        bb.y = Hc[col * ASTR + ks + 2 * half + 1];
        acc[j] = wmma_f32(a, bb, acc[j]);
      }
    }
    __syncthreads();   // everyone done reading buf p before it is refilled
    p ^= 1;
  }

  // ---- epilogue: dst = ALPHA*x + acc (D-matrix VGPR layout) ----
  #pragma unroll
  for (int j = 0; j < 4; ++j) {
    const int col = colw + j * 16 + l16;
    const int cl  = col >> 5;
    const int t   = col & 31;
    #pragma unroll
    for (int r = 0; r < 8; ++r) {
      const int v = v0 + msub * 16 + half * 8 + r;
      const size_t idx = ((size_t)(bc + cl) * NN + v) * TT + t;
      dst[idx] = ALPHA * x[idx] + acc[j][r];
    }
  }
}

// ---------------------------------------------------------------------------
// Kernel 4: y[b,o,n,t] = sigmoid( sum_{c<96} w[o,c]*ho_c[b,n,t] + bias[o] )
// where ho = concat(x, h1, h2). WMMA GEMM: M=o(32), K=96 (3 sources x 32),
// cols = 4 n x 32 t = 128 per WG.
// ---------------------------------------------------------------------------
#define WSTR 100
__global__ __launch_bounds__(256) void MixProp_out(const float* __restrict__ x,
                                                   const float* __restrict__ h1,
                                                   const float* __restrict__ h2,
                                                   const float* __restrict__ w,
                                                   const float* __restrict__ bias,
                                                   float* __restrict__ out) {
  __shared__ float Ws[COUT * WSTR];                // [o][k], k<96
  __shared__ __align__(16) float S[128 * ASTR];    // [col][c], c<32 (one source)

  const int tid  = threadIdx.x;
  const int lane = tid & 31;
  const int wid  = tid >> 5;
  const int l16  = lane & 15;
  const int half = lane >> 4;

  const int n0 = blockIdx.x * 4;
  const int b  = blockIdx.y;

  // stage weights [32 x 96] once
  for (int e = tid; e < COUT * KTOT; e += 256)
    Ws[(e / KTOT) * WSTR + (e % KTOT)] = w[e];

  const int msub = wid & 1;            // o subtile: rows 0-15 / 16-31
  const int colw = (wid >> 1) * 32;    // 32 cols per wave (2 tiles)

  v8f acc[2];
  acc[0] = (v8f){0,0,0,0,0,0,0,0};
  acc[1] = (v8f){0,0,0,0,0,0,0,0};

  const int tld = tid & 31;
  const int kk  = tid >> 5;
  __syncthreads();

  for (int s = 0; s < 3; ++s) {
    const float* src = (s == 0) ? x : ((s == 1) ? h1 : h2);
    // stage 32(c) x 128(col) tile of this source
    #pragma unroll
    for (int nl = 0; nl < 4; ++nl) {
      #pragma unroll
      for (int i = 0; i < 4; ++i) {
        const int c = kk + (i << 3);
        S[(nl * 32 + tld) * ASTR + c] =
            src[((size_t)(b * CC + c) * NN + n0 + nl) * TT + tld];
      }
    }
    __syncthreads();
    #pragma unroll
    for (int ks = 0; ks < 32; ks += 4) {
      const int kg = s * 32 + ks;
      v2f a;
      a.x = Ws[(msub * 16 + l16) * WSTR + kg + 2 * half];
      a.y = Ws[(msub * 16 + l16) * WSTR + kg + 2 * half + 1];
      #pragma unroll
      for (int j = 0; j < 2; ++j) {
        const int col = colw + j * 16 + l16;
        v2f bb;
        bb.x = S[col * ASTR + ks + 2 * half];
        bb.y = S[col * ASTR + ks + 2 * half + 1];
        acc[j] = wmma_f32(a, bb, acc[j]);
      }
    }
    __syncthreads();
  }

  // epilogue: bias + sigmoid, scatter per D layout
  #pragma unroll
  for (int j = 0; j < 2; ++j) {
    const int col = colw + j * 16 + l16;
    const int nl  = col >> 5;
    const int t   = col & 31;
    #pragma unroll
    for (int r = 0; r < 8; ++r) {
      const int o = msub * 16 + half * 8 + r;
      float v = acc[j][r] + bias[o];
      v = 1.0f / (1.0f + __expf(-v));
      out[((size_t)(b * COUT + o) * NN + n0 + nl) * TT + t] = v;
    }
  }
}

// ---------------------------------------------------------------------------
extern "C" void kernel_launch(void* const* d_in, const int* in_sizes, int n_in,
                              void* d_out, int out_size, void* d_ws, size_t ws_size,
                              hipStream_t stream) {
  const float* x    = (const float*)d_in[0];   // [B,C,N,T]
  const float* adj  = (const float*)d_in[1];   // [N,N]
  const float* w    = (const float*)d_in[2];   // [COUT, 96]
  const float* bias = (const float*)d_in[3];   // [COUT]
  float* out = (float*)d_out;

  const size_t XSZ = (size_t)BB * CC * NN * TT;   // 33,554,432 floats
  float* wsf   = (float*)d_ws;
  float* invd  = wsf;                        // 4096 floats
  float* anorm = wsf + 4096;                 // NN*NN floats (16B aligned)
  float* h1    = anorm + (size_t)NN * NN;    // XSZ floats
  float* h2    = h1 + XSZ;                   // XSZ floats

  // 1) degree reciprocals, 2) normalized adjacency (adj+I)*diag(invd)
  MixProp_deg<<<NN, 256, 0, stream>>>(adj, invd);
  MixProp_norm<<<(NN * NN / 4) / 256, 256, 0, stream>>>(adj, invd, anorm);

  // 3) h1 = ALPHA*x + Anorm @ x ; h2 = ALPHA*x + Anorm @ h1
  dim3 ggrid(NN / 64, BB, CC / 4);
  MixProp_gemm<<<ggrid, 256, 0, stream>>>(anorm, x,  x, h1);
  MixProp_gemm<<<ggrid, 256, 0, stream>>>(anorm, h1, x, h2);

  // 4) fused 1x1 conv (96 -> 32) + bias + sigmoid
  dim3 ogrid(NN / 4, BB);
  MixProp_out<<<ogrid, 256, 0, stream>>>(x, h1, h2, w, bias, out);
}